// Post_layer_66992899883345
// MI455X (gfx1250) — compile-verified
//
#include <hip/hip_runtime.h>
#include <hip/hip_bf16.h>

#ifndef __has_builtin
#define __has_builtin(x) 0
#endif

#define NT   256
#define CH   16
#define TILE (NT * CH)        // 4096
#define HALO 12
#define LW   (TILE + 2 * HALO) // 4120
#define LROW 250000

#if __has_builtin(__builtin_amdgcn_global_load_async_to_lds_b128)
#define HAVE_ASYNC_LDS 1
#else
#define HAVE_ASYNC_LDS 0
#endif

typedef int v4i __attribute__((ext_vector_type(4)));
typedef __attribute__((address_space(1))) v4i* gptr_v4i;
typedef __attribute__((address_space(3))) v4i* lptr_v4i;

// LDS swizzle: insert a 4-float pad every 64 floats.
// Keeps 16B alignment for b128 accesses (i % 4 == 0 -> m % 4 == 0) and
// spreads the per-thread 16-float-strided read bases across banks.
__device__ __forceinline__ int swzi(int i) { return i + ((i >> 6) << 2); }

__device__ __forceinline__ int clampi(int v, int lo, int hi) {
    return v < lo ? lo : (v > hi ? hi : v);
}

__device__ __forceinline__ void lds_copy_b128(float* lds_dst, const float* g_src) {
#if HAVE_ASYNC_LDS
    __builtin_amdgcn_global_load_async_to_lds_b128(
        (gptr_v4i)g_src,
        (lptr_v4i)lds_dst,
        0, 0);
#else
    *(float4*)lds_dst = *(const float4*)g_src;
#endif
}

__device__ __forceinline__ void wait_async0() {
#if HAVE_ASYNC_LDS
#if __has_builtin(__builtin_amdgcn_s_wait_asynccnt)
    __builtin_amdgcn_s_wait_asynccnt(0);
#else
    asm volatile("s_wait_asynccnt 0" ::: "memory");
#endif
#endif
}

__global__ __launch_bounds__(NT)
void morph_main(const float* __restrict__ x, float* __restrict__ out, int L) {
    __shared__ float s[LW + (((LW >> 6) + 1) << 2)];

    const int tile  = blockIdx.x;
    const int row   = blockIdx.y;
    const int tbase = tile * TILE;
    const int tid   = threadIdx.x;
    const float* xr = x + (size_t)row * L;
    float* orow     = out + (size_t)row * L;

    const int lo = tbase - HALO;
    const bool interior = (lo >= 0) && (tbase + TILE + HALO <= L);

    if (interior) {
        // 16B-aligned: lo = tile*4096 - 12 (mult of 4), row*L mult of 4.
        for (int v = tid; v < LW / 4; v += NT) {
            const int i = 4 * v;
            lds_copy_b128(&s[swzi(i)], xr + lo + i);
        }
        wait_async0();
    } else {
        for (int i = tid; i < LW; i += NT) {
            const int g = clampi(lo + i, 0, L - 1);
            s[swzi(i)] = xr[g];
        }
    }
    __syncthreads();

    // ---- per-thread register window: 40 inputs -> 16 outputs ----
    const int base = tid * CH; // LDS element index of virtual position c = -12
    float a[40];
#pragma unroll
    for (int k = 0; k < 10; ++k) {
        const int i = base + 4 * k;
        const float4 q = *(const float4*)&s[swzi(i)];
        a[4 * k + 0] = q.x; a[4 * k + 1] = q.y;
        a[4 * k + 2] = q.z; a[4 * k + 3] = q.w;
    }

    // Stage 1: dilation k=9 -> s1[i], i=0..31 (c = i-8), via log-step maxes
    float w2[38];
#pragma unroll
    for (int i = 0; i < 38; ++i) w2[i] = fmaxf(a[i], a[i + 1]);
    float w4[36];
#pragma unroll
    for (int i = 0; i < 36; ++i) w4[i] = fmaxf(w2[i], w2[i + 2]);
    float w8[32];
#pragma unroll
    for (int i = 0; i < 32; ++i) w8[i] = fmaxf(w4[i], w4[i + 4]);
    float s1[32];
#pragma unroll
    for (int i = 0; i < 32; ++i) s1[i] = fmaxf(w8[i], a[i + 8]);

    // Stage 2: erosion k=9 -> s2[j], j=0..23 (c = j-4)
    float e2[30];
#pragma unroll
    for (int i = 0; i < 30; ++i) e2[i] = fminf(s1[i], s1[i + 1]);
    float e4[28];
#pragma unroll
    for (int i = 0; i < 28; ++i) e4[i] = fminf(e2[i], e2[i + 2]);
    float e8[24];
#pragma unroll
    for (int i = 0; i < 24; ++i) e8[i] = fminf(e4[i], e4[i + 4]);
    float s2[24];
#pragma unroll
    for (int i = 0; i < 24; ++i) s2[i] = fminf(e8[i], s1[i + 8]);

    // Stage 3: erosion k=5 -> s3[m], m=0..19 (c = m-2)
    float f2[22];
#pragma unroll
    for (int i = 0; i < 22; ++i) f2[i] = fminf(s2[i], s2[i + 1]);
    float f4[20];
#pragma unroll
    for (int i = 0; i < 20; ++i) f4[i] = fminf(f2[i], f2[i + 2]);
    float s3[20];
#pragma unroll
    for (int i = 0; i < 20; ++i) s3[i] = fminf(f4[i], s2[i + 4]);

    // Stage 4: dilation k=5 -> o[n], n=0..15 (c = n)
    float g2[18];
#pragma unroll
    for (int i = 0; i < 18; ++i) g2[i] = fmaxf(s3[i], s3[i + 1]);
    float g4[16];
#pragma unroll
    for (int i = 0; i < 16; ++i) g4[i] = fmaxf(g2[i], g2[i + 2]);
    float o[16];
#pragma unroll
    for (int i = 0; i < 16; ++i) o[i] = fmaxf(g4[i], s3[i + 4]);

    const int obase = tbase + base;
    if (obase + CH <= L) {
        float4* o4 = (float4*)(orow + obase);
        o4[0] = make_float4(o[0],  o[1],  o[2],  o[3]);
        o4[1] = make_float4(o[4],  o[5],  o[6],  o[7]);
        o4[2] = make_float4(o[8],  o[9],  o[10], o[11]);
        o4[3] = make_float4(o[12], o[13], o[14], o[15]);
    } else {
#pragma unroll
        for (int i = 0; i < CH; ++i)
            if (obase + i < L) orow[obase + i] = o[i];
    }
}

// Exact per-stage replicate padding at row ends (first/last 16 outputs per row).
// Overwrites the main kernel's approximate edge values; ordered on the stream.
__global__ __launch_bounds__(32)
void morph_fixup(const float* __restrict__ x, float* __restrict__ out, int L) {
    const int row = blockIdx.x;
    const int t   = threadIdx.x;            // 0..31
    const int p   = (t < 16) ? t : (L - 32 + t);
    const float* xr = x + (size_t)row * L;

    const float NINF = -__builtin_inff();
    const float PINF =  __builtin_inff();

    float S1[17];
    for (int j = 0; j < 17; ++j) {
        const int q = p - 8 + j;
        float m = NINF;
        for (int d = -4; d <= 4; ++d)
            m = fmaxf(m, xr[clampi(q + d, 0, L - 1)]);
        S1[j] = m;
    }
    float S2[9];
    for (int j = 0; j < 9; ++j) {
        const int r = p - 4 + j;
        float m = PINF;
        for (int d = -4; d <= 4; ++d) {
            const int q = clampi(r + d, 0, L - 1);
            m = fminf(m, S1[q - (p - 8)]);
        }
        S2[j] = m;
    }
    float S3[5];
    for (int j = 0; j < 5; ++j) {
        const int u = p - 2 + j;
        float m = PINF;
        for (int d = -2; d <= 2; ++d) {
            const int r = clampi(u + d, 0, L - 1);
            m = fminf(m, S2[r - (p - 4)]);
        }
        S3[j] = m;
    }
    float m = NINF;
    for (int d = -2; d <= 2; ++d) {
        const int u = clampi(p + d, 0, L - 1);
        m = fmaxf(m, S3[u - (p - 2)]);
    }
    out[(size_t)row * L + p] = m;
}

extern "C" void kernel_launch(void* const* d_in, const int* in_sizes, int n_in,
                              void* d_out, int out_size, void* d_ws, size_t ws_size,
                              hipStream_t stream) {
    (void)n_in; (void)out_size; (void)d_ws; (void)ws_size;
    const float* x = (const float*)d_in[0];
    float* out     = (float*)d_out;
    const int L    = LROW;
    const int rows = in_sizes[0] / L;                 // 32*12 = 384
    const int tiles = (L + TILE - 1) / TILE;          // 62

    dim3 grid(tiles, rows);
    morph_main<<<grid, NT, 0, stream>>>(x, out, L);
    morph_fixup<<<dim3(rows), 32, 0, stream>>>(x, out, L);
}